// WindowedSelfAttention_46256797778383
// MI455X (gfx1250) — compile-verified
//
#include <hip/hip_runtime.h>
#include <hip/hip_bf16.h>

typedef float v2f __attribute__((ext_vector_type(2)));
typedef float v4f __attribute__((ext_vector_type(4)));
typedef float v8f __attribute__((ext_vector_type(8)));

#define E_ 512
#define H_ 8
#define D_ 64
#define WIN_ 256
#define N_ 4
#define S_ 2048
#define M_TOK (N_ * S_)          // 8192 token rows
#define BAND 272                 // max band width: 256 + 16 (16 query rows share band)
#define BSTRIDE 274              // LDS row stride (even, avoids systematic bank conflicts)

// ---------------------------------------------------------------------------
// Projection GEMM: Y = X @ W.T + b  (X:[8192,512], W:[512,512] row=out)
// Each wave computes a 16x16 tile with 128 V_WMMA_F32_16X16X4_F32 ops.
// f32 WMMA layouts (ISA 7.12.2): A 16x4 & B 4x16: k = vgpr + 2*(lane>=16),
// row/col = lane&15  ->  each lane loads a consecutive float2 per operand.
// out_mode 0: plain [M,512];  out_mode 1: remap to head-major [N,H,S,D].
// ---------------------------------------------------------------------------
__global__ __launch_bounds__(256) void proj_gemm_f32(
    const float* __restrict__ X, const float* __restrict__ W,
    const float* __restrict__ bias, float* __restrict__ Y, int out_mode) {
  const int lane = threadIdx.x & 31;
  const int wid  = threadIdx.x >> 5;          // 0..7
  const int wr   = wid >> 2, wc = wid & 3;    // 2x4 wave grid
  const int m0   = blockIdx.x * 32 + wr * 16; // token row base
  const int o0   = blockIdx.y * 64 + wc * 16; // output-feature base
  const int hi   = lane >> 4;                 // lane half selects k+2
  const int ln   = lane & 15;

  const float* xrow = X + (size_t)(m0 + ln) * E_ + hi * 2;
  const float* wrow = W + (size_t)(o0 + ln) * E_ + hi * 2;

  v8f acc = {};
  for (int e = 0; e < E_; e += 4) {
    v2f a = *(const v2f*)(xrow + e);   // A[m=ln, k=e+2*hi .. +1]
    v2f b = *(const v2f*)(wrow + e);   // B[k, n=ln] = W[o0+ln, e+k]
    acc = __builtin_amdgcn_wmma_f32_16x16x4_f32(false, a, false, b,
                                                (short)0, acc, false, false);
  }

  const int o = o0 + ln;
  const float bv = bias[o];
  for (int r = 0; r < 8; ++r) {                // D: m = r + 8*hi, n = ln
    const int m = m0 + r + hi * 8;
    const float val = acc[r] + bv;
    if (out_mode == 0) {
      Y[(size_t)m * E_ + o] = val;
    } else {
      const int nb = m >> 11, s = m & (S_ - 1);
      const int hh = o >> 6, dd = o & (D_ - 1);
      Y[(((size_t)nb * H_ + hh) * S_ + s) * D_ + dd] = val;
    }
  }
}

// ---------------------------------------------------------------------------
// Per-(n,h) column sums of V: vtot[n,h,d] = sum_s v[n,h,s,d]
// ---------------------------------------------------------------------------
__global__ void vtot_kernel(const float* __restrict__ vh, float* __restrict__ vtot) {
  const int idx = blockIdx.x * blockDim.x + threadIdx.x;   // N*H*D = 2048
  if (idx >= N_ * H_ * D_) return;
  const int nh = idx / D_, d = idx % D_;
  const float* p = vh + (size_t)nh * S_ * D_ + d;
  float s = 0.0f;
  for (int j = 0; j < S_; ++j) s += p[(size_t)j * D_];
  vtot[idx] = s;
}

// ---------------------------------------------------------------------------
// Windowed attention. One block = (n, h, 16 query rows), 128 threads (4 waves).
// Out-of-window logits are 0 in the reference -> uniform tail u = e^{-m}/Z.
// ctx_i = sum_{j in band} (a_ij - u_i) v_j  +  u_i * vtot     (exact)
// ---------------------------------------------------------------------------
__global__ __launch_bounds__(128) void attn_kernel(
    const float* __restrict__ qh, const float* __restrict__ kh,
    const float* __restrict__ vh, const float* __restrict__ vtot,
    const int* __restrict__ mask, float* __restrict__ att_out,
    float* __restrict__ ctx) {
  __shared__ __align__(16) float s_lds[16 * BSTRIDE];
  __shared__ float row_m[16], row_invZ[16], row_u[16];
  __shared__ float red[16][8];

  const int q0   = blockIdx.x * 16;
  const int h    = blockIdx.y;
  const int n    = blockIdx.z;
  const int lane = threadIdx.x & 31;
  const int wid  = threadIdx.x >> 5;    // 0..3
  const int hi   = lane >> 4;
  const int ln   = lane & 15;
  const size_t nhS = ((size_t)n * H_ + h) * S_;

  const int jb0 = max(0, q0 - WIN_ / 2);
  const int jb1 = min(S_ - 1, q0 + 15 + WIN_ / 2);
  const int wb  = jb1 - jb0 + 1;                 // <= 272
  const int ntiles = (wb + 15) / 16;
  const float scale = 0.044194173824159216f;     // 1/sqrt(512)
  const int* mrow = mask + n * S_;

  // ---- Phase 1: band scores S = Q K^T via WMMA f32, tiles over 4 waves ----
  const float* qbase = qh + (nhS + q0 + ln) * D_ + hi * 2;
  for (int t = wid; t < ntiles; t += 4) {
    const int jg = min(jb0 + t * 16 + ln, S_ - 1);   // clamp edge garbage
    const float* kbase = kh + (nhS + jg) * D_ + hi * 2;
    v8f acc = {};
    for (int dc = 0; dc < D_; dc += 4) {
      v2f a = *(const v2f*)(qbase + dc);
      v2f b = *(const v2f*)(kbase + dc);
      acc = __builtin_amdgcn_wmma_f32_16x16x4_f32(false, a, false, b,
                                                  (short)0, acc, false, false);
    }
    const int jl  = t * 16 + ln;
    const int jgl = jb0 + jl;
    const bool live = (jgl <= jb1) && (mrow[min(jgl, S_ - 1)] != 0);
    for (int r = 0; r < 8; ++r) {
      const int m = r + hi * 8;
      s_lds[m * BSTRIDE + jl] = live ? acc[r] * scale : -1e30f;
    }
  }
  __syncthreads();

  // ---- Phase 2: softmax stats (8 threads/row, LDS tree reduction) ----
  {
    const int i  = threadIdx.x >> 3;               // row 0..15
    const int t  = threadIdx.x & 7;                // sub-lane 0..7
    const int qi = q0 + i;
    const int wi0 = max(0, qi - WIN_ / 2), wi1 = min(S_ - 1, qi + WIN_ / 2);

    float mmax = 0.0f;                             // tail logit 0 always present
    for (int j = wi0 + t; j <= wi1; j += 8)
      mmax = fmaxf(mmax, s_lds[i * BSTRIDE + (j - jb0)]);
    red[i][t] = mmax;
    __syncthreads();
    if (t == 0) {
      float m2 = red[i][0];
      for (int r = 1; r < 8; ++r) m2 = fmaxf(m2, red[i][r]);
      row_m[i] = m2;
    }
    __syncthreads();

    const float mi = row_m[i];
    float zs = 0.0f;
    for (int j = wi0 + t; j <= wi1; j += 8)
      zs += __expf(s_lds[i * BSTRIDE + (j - jb0)] - mi);
    red[i][t] = zs;
    __syncthreads();
    if (t == 0) {
      float z2 = 0.0f;
      for (int r = 0; r < 8; ++r) z2 += red[i][r];
      const float tail = __expf(-mi);
      z2 += (float)(S_ - (wi1 - wi0 + 1)) * tail;  // uniform zero-logit tail
      const float invZ = 1.0f / z2;
      row_invZ[i] = invZ;
      row_u[i]    = tail * invZ;
    }
  }
  __syncthreads();

  // ---- Phase 3: stream full 2048-wide attention rows (dominant traffic) ----
  // float4 non-temporal stores: global_store_b128 + NT hint so this one-shot
  // 537MB stream doesn't evict the L2-resident Q/K/V tiles.
  float* arow_base = att_out + (nhS + q0) * (size_t)S_;
  for (int i = 0; i < 16; ++i) {
    const int qi = q0 + i;
    const int wi0 = max(0, qi - WIN_ / 2), wi1 = min(S_ - 1, qi + WIN_ / 2);
    const float mi = row_m[i], invZ = row_invZ[i], u = row_u[i];
    float* arow = arow_base + (size_t)i * S_;
    for (int j4 = threadIdx.x * 4; j4 < S_; j4 += blockDim.x * 4) {
      v4f av;
      for (int c = 0; c < 4; ++c) {
        const int j = j4 + c;
        float a = u;
        if (j >= wi0 && j <= wi1)
          a = __expf(s_lds[i * BSTRIDE + (j - jb0)] - mi) * invZ;
        av[c] = a;
      }
      __builtin_nontemporal_store(av, (v4f*)(arow + j4));
    }
  }
  __syncthreads();

  // ---- Phase 4: overwrite LDS band with coefficients (a - u), 0 outside ----
  for (int idx = threadIdx.x; idx < 16 * BAND; idx += blockDim.x) {
    const int i = idx / BAND, jl = idx % BAND;
    const int qi = q0 + i, jg = jb0 + jl;
    const int wi0 = max(0, qi - WIN_ / 2), wi1 = min(S_ - 1, qi + WIN_ / 2);
    float c = 0.0f;
    if (jg >= wi0 && jg <= wi1)
      c = __expf(s_lds[i * BSTRIDE + jl] - row_m[i]) * row_invZ[i] - row_u[i];
    s_lds[i * BSTRIDE + jl] = c;
  }
  __syncthreads();

  // ---- Phase 5: ctx = coeff[16 x 272] * V[272 x 64] via WMMA (A from LDS) ----
  {
    const int dblk = wid * 16;                       // each wave: one 16-d slab
    const float* vbase = vh + nhS * D_ + dblk + ln;  // + j*D
    v8f acc = {};
    for (int jc = 0; jc < BAND; jc += 4) {
      const int ja = jc + hi * 2;
      v2f a = *(const v2f*)(&s_lds[ln * BSTRIDE + ja]);   // ds_load_b64
      v2f b;
      b.x = vbase[(size_t)min(jb0 + ja,     S_ - 1) * D_];
      b.y = vbase[(size_t)min(jb0 + ja + 1, S_ - 1) * D_];
      acc = __builtin_amdgcn_wmma_f32_16x16x4_f32(false, a, false, b,
                                                  (short)0, acc, false, false);
    }
    const float vtd = vtot[((size_t)n * H_ + h) * D_ + dblk + ln];
    for (int r = 0; r < 8; ++r) {
      const int m  = r + hi * 8;
      const int qi = q0 + m;
      ctx[((size_t)n * S_ + qi) * E_ + h * D_ + dblk + ln] = acc[r] + row_u[m] * vtd;
    }
  }
}

// ---------------------------------------------------------------------------
extern "C" void kernel_launch(void* const* d_in, const int* in_sizes, int n_in,
                              void* d_out, int out_size, void* d_ws, size_t ws_size,
                              hipStream_t stream) {
  (void)in_sizes; (void)n_in; (void)out_size; (void)ws_size;
  const float* value = (const float*)d_in[0];
  const float* key_  = (const float*)d_in[1];
  const float* query = (const float*)d_in[2];
  const int*   mask  = (const int*)d_in[3];
  const float* Wv = (const float*)d_in[4];  const float* bv = (const float*)d_in[5];
  const float* Wk = (const float*)d_in[6];  const float* bk = (const float*)d_in[7];
  const float* Wq = (const float*)d_in[8];  const float* bq = (const float*)d_in[9];
  const float* Wo = (const float*)d_in[10]; const float* bo = (const float*)d_in[11];

  float* out = (float*)d_out;                       // [N,S,E]
  float* att = out + (size_t)N_ * S_ * E_;          // [N,H,S,S]

  const size_t NHSD = (size_t)N_ * H_ * S_ * D_;    // 4.19M floats each
  float* qh   = (float*)d_ws;
  float* kh   = qh + NHSD;
  float* vh   = kh + NHSD;
  float* ctx  = vh + NHSD;                          // [N,S,E]
  float* vtot = ctx + (size_t)N_ * S_ * E_;         // [N,H,D]

  dim3 gproj(M_TOK / 32, E_ / 64);
  proj_gemm_f32<<<gproj, 256, 0, stream>>>(query, Wq, bq, qh, 1);
  proj_gemm_f32<<<gproj, 256, 0, stream>>>(key_,  Wk, bk, kh, 1);
  proj_gemm_f32<<<gproj, 256, 0, stream>>>(value, Wv, bv, vh, 1);
  vtot_kernel<<<(N_ * H_ * D_ + 255) / 256, 256, 0, stream>>>(vh, vtot);
  attn_kernel<<<dim3(S_ / 16, H_, N_), 128, 0, stream>>>(qh, kh, vh, vtot, mask, att, ctx);
  proj_gemm_f32<<<gproj, 256, 0, stream>>>(ctx, Wo, bo, out, 0);
}